// BERT_CRF_Model_7129645711815
// MI455X (gfx1250) — compile-verified
//
#include <hip/hip_runtime.h>

typedef __attribute__((ext_vector_type(2))) float v2f;
typedef __attribute__((ext_vector_type(4))) float v4f;
typedef __attribute__((ext_vector_type(8))) float v8f;

#define Bn 64
#define Sn 512
#define En 1024
#define Tn 9

// ---------------------------------------------------------------------------
// Kernel 0: build zero-padded W^T stored as [16][E] row-major (row t = tag).
// 36KB of real data + zeros; lives in d_ws, L0-cacheable for the GEMM.
// ---------------------------------------------------------------------------
__global__ void wpad_kernel(const float* __restrict__ W, float* __restrict__ WtPad) {
  int i = blockIdx.x * blockDim.x + threadIdx.x;   // 0 .. 16*1024-1
  int t = i >> 10;
  int k = i & 1023;
  WtPad[i] = (t < Tn) ? W[t * En + k] : 0.0f;
}

// ---------------------------------------------------------------------------
// Kernel 1: fused GEMM (f32 WMMA 16x16x4) + log_softmax over 9 classes.
// One wave per 16-row tile, 8 waves / block, 256 blocks -> 32768 rows.
// K permutation: WMMA j of a 16-wide K chunk covers physical
// k in {base+2j, base+2j+1, base+8+2j, base+8+2j+1}; half h supplies
// k = base + 8h + 2j (+1) => per-lane A loads are contiguous b128.
// ---------------------------------------------------------------------------
__global__ __launch_bounds__(256) void gemm_lsm_kernel(
    const float* __restrict__ X,       // (32768, 1024)
    const float* __restrict__ WtPad,   // (16, 1024), rows 9..15 zero
    const float* __restrict__ bias,    // (9)
    float* __restrict__ emission)      // (32768, 9)
{
  __shared__ float tile[8][16][17];    // padded to dodge bank conflicts
  const int wave = threadIdx.x >> 5;
  const int lane = threadIdx.x & 31;
  const int half = lane >> 4;
  const int idx  = lane & 15;
  const long rowBase = (long)blockIdx.x * 128 + wave * 16;

  const float* arow = X + (rowBase + idx) * (long)En;  // A: lane idx = row M
  const float* brow = WtPad + idx * En;                // B: lane idx = col N (tag)

  v8f acc = {};
  for (int base = 0; base < En; base += 16) {
    v4f a0 = *(const v4f*)(arow + base + 8 * half);      // k = base+8h .. +3
    v4f a1 = *(const v4f*)(arow + base + 8 * half + 4);  // k = base+8h+4 .. +7
#pragma unroll
    for (int j = 0; j < 4; ++j) {
      const int k = base + 8 * half + 2 * j;
      v4f aq = (j < 2) ? a0 : a1;
      v2f a;
      a.x = (j & 1) ? aq.z : aq.x;
      a.y = (j & 1) ? aq.w : aq.y;
      v2f b = *(const v2f*)(brow + k);                   // W[idx][k], W[idx][k+1]
      acc = __builtin_amdgcn_wmma_f32_16x16x4_f32(
          false, a, false, b, (short)0, acc, false, false);
    }
  }

  // D layout: vgpr r, lane (h,idx) = tag_space[rowBase + r + 8h][idx]
#pragma unroll
  for (int r = 0; r < 8; ++r)
    tile[wave][r + 8 * half][idx] = acc[r];
  __syncthreads();

  if (lane < 16) {
    const long row = rowBase + lane;
    float x[Tn];
    float mx = -3.0e38f;
#pragma unroll
    for (int t = 0; t < Tn; ++t) {
      x[t] = tile[wave][lane][t] + bias[t];
      mx = fmaxf(mx, x[t]);
    }
    float s = 0.0f;
#pragma unroll
    for (int t = 0; t < Tn; ++t) s += __expf(x[t] - mx);
    const float lse = mx + __logf(s);
    float* o = emission + row * Tn;
#pragma unroll
    for (int t = 0; t < Tn; ++t) o[t] = x[t] - lse;
  }
}

// ---------------------------------------------------------------------------
// Kernel 2: CRF log-likelihood per batch. One wave (32 threads) per batch.
// Lanes 0..8 hold the 9 forward-scan states; __shfl broadcasts scores.
// Numerator parallelized over s across all 32 lanes, fixed-order reduction.
// ---------------------------------------------------------------------------
__global__ __launch_bounds__(32) void crf_kernel(
    const float* __restrict__ emission,  // (B*S, 9)
    const int* __restrict__ tags,        // (B, S)
    const int* __restrict__ mask,        // (B, S)
    const float* __restrict__ start_tr,  // (9)
    const float* __restrict__ end_tr,    // (9)
    const float* __restrict__ trans,     // (9, 9)
    float* __restrict__ partial)         // (B)
{
  const int b = blockIdx.x;
  const int lane = threadIdx.x;
  const float* em = emission + (long)b * Sn * Tn;
  const int* tg = tags + b * Sn;
  const int* mk = mask + b * Sn;

  // ---- numerator ----
  float nsum = 0.0f;
  int msum = 0;
  for (int s = lane; s < Sn; s += 32) {
    msum += mk[s];
    if (s >= 1) {
      const float mf = (float)mk[s];
      const int tp = tg[s - 1], tc = tg[s];
      nsum += mf * (trans[tp * Tn + tc] + em[s * Tn + tc]);
    }
  }
#pragma unroll
  for (int off = 16; off >= 1; off >>= 1) {
    nsum += __shfl_down(nsum, off, 32);
    msum += __shfl_down(msum, off, 32);
  }
  nsum = __shfl(nsum, 0, 32);
  msum = __shfl(msum, 0, 32);
  const int seq_last = msum - 1;
  const int t0 = tg[0];
  const int tl = tg[seq_last];
  const float numerator = start_tr[t0] + em[t0] + nsum + end_tr[tl];

  // ---- denominator: forward algorithm ----
  const int j = lane;
  const bool valid = (j < Tn);
  float tcol[Tn];
#pragma unroll
  for (int i = 0; i < Tn; ++i) tcol[i] = valid ? trans[i * Tn + j] : 0.0f;
  float sc = valid ? (start_tr[j] + em[j]) : -1.0e30f;

  for (int s = 1; s < Sn; ++s) {
    const float e = valid ? em[s * Tn + j] : 0.0f;
    const int m = mk[s];
    float v[Tn];
    float mx = -3.0e38f;
#pragma unroll
    for (int i = 0; i < Tn; ++i) {
      v[i] = __shfl(sc, i, 32) + tcol[i];
      mx = fmaxf(mx, v[i]);
    }
    float sm = 0.0f;
#pragma unroll
    for (int i = 0; i < Tn; ++i) sm += __expf(v[i] - mx);
    const float nxt = e + mx + __logf(sm);
    sc = (m > 0 && valid) ? nxt : sc;
  }

  const float y = valid ? (sc + end_tr[j]) : -1.0e30f;
  float mx2 = -3.0e38f;
  float yy[Tn];
#pragma unroll
  for (int i = 0; i < Tn; ++i) {
    yy[i] = __shfl(y, i, 32);
    mx2 = fmaxf(mx2, yy[i]);
  }
  float s2 = 0.0f;
#pragma unroll
  for (int i = 0; i < Tn; ++i) s2 += __expf(yy[i] - mx2);
  const float den = mx2 + __logf(s2);

  if (lane == 0) partial[b] = numerator - den;
}

// ---------------------------------------------------------------------------
// Kernel 3: deterministic serial sum of 64 partials (no float atomics).
// ---------------------------------------------------------------------------
__global__ void final_reduce(const float* __restrict__ partial,
                             float* __restrict__ out) {
  if (threadIdx.x == 0 && blockIdx.x == 0) {
    float s = 0.0f;
    for (int i = 0; i < Bn; ++i) s += partial[i];
    out[0] = s;
  }
}

extern "C" void kernel_launch(void* const* d_in, const int* in_sizes, int n_in,
                              void* d_out, int out_size, void* d_ws, size_t ws_size,
                              hipStream_t stream) {
  const float* X     = (const float*)d_in[0];  // pretraining_embed (64,512,1024)
  const int*   tags  = (const int*)d_in[1];    // (64,512)
  const int*   mask  = (const int*)d_in[2];    // (64,512)
  const float* W     = (const float*)d_in[3];  // (9,1024)
  const float* bias  = (const float*)d_in[4];  // (9)
  const float* st    = (const float*)d_in[5];  // (9)
  const float* en    = (const float*)d_in[6];  // (9)
  const float* tr    = (const float*)d_in[7];  // (9,9)

  float* out      = (float*)d_out;
  float* emission = out + 1;                   // (32768, 9)
  float* wtpad    = (float*)d_ws;              // 16*1024 floats
  float* partial  = wtpad + 16 * En;           // 64 floats

  wpad_kernel<<<64, 256, 0, stream>>>(W, wtpad);
  gemm_lsm_kernel<<<256, 256, 0, stream>>>(X, wtpad, bias, emission);
  crf_kernel<<<Bn, 32, 0, stream>>>(emission, tags, mask, st, en, tr, partial);
  final_reduce<<<1, 32, 0, stream>>>(partial, out);
}